// FastSegmentedPolynomialFromUniform1dJit_9062380994708
// MI455X (gfx1250) — compile-verified
//
#include <hip/hip_runtime.h>

// ---------------------------------------------------------------------------
// Segmented polynomial: out[b, i2[p]*128+e] += coeffs[p]*x[b,i0[p]*128+e]*y[b,i1[p]*128+e]
// B=16384, E=128, S0=S1=8, S2=16, P=64.
//
// Memory-bound: 256 MB total traffic -> ~11us at 23.3 TB/s. Strategy:
//  * stage x/y rows (4KB each) in LDS via async global->LDS loads (double buffered)
//  * express the 64-path contraction per 16-element e-tile as a
//    16x16x64 GEMM = 16 chained V_WMMA_F32_16X16X4_F32 (wave32 layout)
//  * transpose result through padded LDS so output stores are coalesced B128
// ---------------------------------------------------------------------------

#define EE   128
#define S0N  8
#define S2N  16
#define BPW  16        // batches per workgroup

typedef __attribute__((ext_vector_type(2))) float v2f;
typedef __attribute__((ext_vector_type(8))) float v8f;
typedef __attribute__((ext_vector_type(4))) int   v4i;

typedef __attribute__((address_space(1))) v4i* gv4i_p;   // global int4*
typedef __attribute__((address_space(3))) v4i* lv4i_p;   // LDS int4*

#if __has_builtin(__builtin_amdgcn_global_load_async_to_lds_b128)
#define HAVE_ASYNC 1
#else
#define HAVE_ASYNC 0
#endif

// LDS layout (floats):
#define XBUF0 0          // x row, buffer 0   (1024 floats)
#define YBUF0 1024       // y row, buffer 0
#define XBUF1 2048       // x row, buffer 1
#define YBUF1 3072       // y row, buffer 1
#define OUTB  4096       // out staging: 16 segments, stride 132 (pad kills bank conflicts)
#define OSTR  132
#define NLDS  (OUTB + 16*OSTR)

__global__ __launch_bounds__(256) void seg_poly_wmma(
    const float* __restrict__ gx, const float* __restrict__ gy,
    const float* __restrict__ gcoef, const int* __restrict__ gpath,
    float* __restrict__ gout, int B)
{
  __shared__ __align__(16) float sm[NLDS];

  const int tid  = threadIdx.x;
  const int lane = tid & 31;
  const int wave = tid >> 5;       // 8 waves; wave w owns e in [16w, 16w+16)
  const int row  = lane & 15;      // A-matrix: M row (= local e); B-matrix: N col (= segment)
  const int kh   = lane >> 4;      // K half within a 16x16x4 step
  const int eb   = wave * 16;

  // ---- per-lane precompute: W fragments (constant B-matrix) and z-gather LDS indices
  // K for step s, vgpr j, lane-half kh:  kg = 4*s + 2*kh + j   (ISA 7.12.2 f32 layouts)
  float wf0[16], wf1[16];
  int xi0[16], xi1[16], yi0[16], yi1[16];
#pragma unroll
  for (int s = 0; s < 16; ++s) {
    const int k0 = 4*s + 2*kh;
    const int k1 = k0 + 1;
    const int a0 = gpath[3*k0+0], b0 = gpath[3*k0+1], c0 = gpath[3*k0+2];
    const int a1 = gpath[3*k1+0], b1 = gpath[3*k1+1], c1 = gpath[3*k1+2];
    wf0[s] = (c0 == row) ? gcoef[k0] : 0.0f;   // W[k, n] = coeffs[k] * (i2[k]==n)
    wf1[s] = (c1 == row) ? gcoef[k1] : 0.0f;
    xi0[s] = a0*EE + eb + row;                 // LDS float index of x[i0[k], e]
    xi1[s] = a1*EE + eb + row;
    yi0[s] = b0*EE + eb + row;
    yi1[s] = b1*EE + eb + row;
  }

  const int b0g = blockIdx.x * BPW;

  // ---- async global -> LDS row copy: 256 lanes * 16B = one full 4KB row per instr
  auto prefetch = [&](int xb, int yb, int b) {
    const float* xs = gx + (size_t)b * (S0N*EE) + tid*4;
    const float* ys = gy + (size_t)b * (S0N*EE) + tid*4;
#if HAVE_ASYNC
    __builtin_amdgcn_global_load_async_to_lds_b128(
        (gv4i_p)(v4i*)xs,
        (lv4i_p)(v4i*)&sm[xb + tid*4], 0, 0);
    __builtin_amdgcn_global_load_async_to_lds_b128(
        (gv4i_p)(v4i*)ys,
        (lv4i_p)(v4i*)&sm[yb + tid*4], 0, 0);
#else
    const float4 vx = *(const float4*)xs;
    const float4 vy = *(const float4*)ys;
    *(float4*)&sm[xb + tid*4] = vx;
    *(float4*)&sm[yb + tid*4] = vy;
#endif
  };

  auto body = [&](int xb, int yb, int b) {
    if (b >= B) b = B - 1;   // tail safety (B%16==0 in practice; writes stay deterministic)
#if HAVE_ASYNC
    // allow the other buffer's 2 prefetch loads to stay in flight
#if __has_builtin(__builtin_amdgcn_s_wait_asynccnt)
    __builtin_amdgcn_s_wait_asynccnt(2);
#else
    asm volatile("s_wait_asynccnt 0x2" ::: "memory");
#endif
#endif
    __syncthreads();   // current x/y buffer visible to all waves; out staging reusable

    // ---- 16x16(x64) GEMM over paths: D[e, s2] = sum_k z[e,k] * W[k, s2]
    v8f acc = {0.f,0.f,0.f,0.f,0.f,0.f,0.f,0.f};
#pragma unroll
    for (int s = 0; s < 16; ++s) {
      v2f a, w;
      a[0] = sm[xb + xi0[s]] * sm[yb + yi0[s]];
      a[1] = sm[xb + xi1[s]] * sm[yb + yi1[s]];
      w[0] = wf0[s];
      w[1] = wf1[s];
      acc = __builtin_amdgcn_wmma_f32_16x16x4_f32(
          /*neg_a=*/false, a, /*neg_b=*/false, w,
          /*c_mod=*/(short)0, acc, /*reuse_a=*/false, /*reuse_b=*/false);
    }

    // ---- D tile -> padded LDS staging (transpose e-major -> segment-major)
    // D layout: vgpr r, lanes 0-15: M=r, N=lane; lanes 16-31: M=8+r, N=lane-16
#pragma unroll
    for (int r = 0; r < 8; ++r)
      sm[OUTB + row*OSTR + eb + 8*kh + r] = acc[r];

    __syncthreads();   // all waves done reading x/y buffer + out staging complete

    // prefetch batch b+2 into the buffer we just finished reading
    int bn = b + 2; if (bn >= B) bn = B - 1;
    prefetch(xb, yb, bn);

    // ---- coalesced stream-out: 2048 floats, 256 threads * 2 x float4
    const size_t ob = (size_t)b * (S2N*EE);
#pragma unroll
    for (int c = 0; c < 2; ++c) {
      const int f = c*1024 + tid*4;            // float offset within out row
      const int n = f >> 7;                    // segment
      const int m = f & 127;                   // element
      const float4 v = *(const float4*)&sm[OUTB + n*OSTR + m];
      *(float4*)(gout + ob + f) = v;
    }
  };

  prefetch(XBUF0, YBUF0, b0g);
  prefetch(XBUF1, YBUF1, (b0g + 1 < B) ? b0g + 1 : B - 1);
  for (int i = 0; i < BPW; i += 2) {
    body(XBUF0, YBUF0, b0g + i);
    body(XBUF1, YBUF1, b0g + i + 1);
  }
}

extern "C" void kernel_launch(void* const* d_in, const int* in_sizes, int n_in,
                              void* d_out, int out_size, void* d_ws, size_t ws_size,
                              hipStream_t stream) {
  (void)n_in; (void)d_ws; (void)ws_size; (void)out_size;
  const float* x  = (const float*)d_in[0];
  const float* y  = (const float*)d_in[1];
  const float* cf = (const float*)d_in[2];
  const int*   pi = (const int*)d_in[3];
  float* out = (float*)d_out;
  const int B = in_sizes[0] / (S0N * EE);       // 16384
  const int grid = (B + BPW - 1) / BPW;         // 1024 workgroups of 256 threads
  seg_poly_wmma<<<grid, 256, 0, stream>>>(x, y, cf, pi, out, B);
}